// CharGRU2_86990267613904
// MI455X (gfx1250) — compile-verified
//
#include <hip/hip_runtime.h>

typedef __attribute__((ext_vector_type(2))) float v2f;
typedef __attribute__((ext_vector_type(8))) float v8f;

namespace {
constexpr int kB  = 2048;   // batch
constexpr int kT  = 128;    // seq len
constexpr int kV  = 256;    // vocab
constexpr int kH  = 20;     // hidden
constexpr int kH3 = 60;     // 3*H gate width
constexpr int kL  = 15;     // labels
constexpr int kNT = 4;      // N tiles of 16 (60 -> 64 padded)
constexpr int kNK = 5;      // K steps of 4 (K = 20 exactly)

// padded LDS strides (bank-conflict-free mod 64 banks)
constexpr int kXP = 129;    // token-id row stride (129 % 64 == 1)
constexpr int kHP = 22;     // hidden-state row stride (even -> b64-aligned A-frags)
constexpr int kRP = 66;     // matmul-dump row stride

// dynamic LDS layout (bytes)
constexpr int kW0Off  = 0;                          // 256*60 f32 (b0i folded) = 61440 B
constexpr int kXOff   = kW0Off + kV * kH3 * 4;      // 16*129 i32             =  8256 B
constexpr int kH0Off  = kXOff + 16 * kXP * 4;       // 16*22 f32
constexpr int kH1Off  = kH0Off + 16 * kHP * 4;      // 16*22 f32
constexpr int kRecOff = kH1Off + 16 * kHP * 4;      // 16*66 f32
constexpr int kXwOff  = kRecOff + 16 * kRP * 4;     // 16*66 f32
constexpr int kW1fOff = kXwOff + 16 * kRP * 4;      // 20 frags * 32 lanes * 8B = 5120 B
constexpr int kU1fOff = kW1fOff + kNT * kNK * 32 * 8;
constexpr int kLdsBytes = kU1fOff + kNT * kNK * 32 * 8;  // = 91200 B (< 320 KB/WGP)
}

#define WMMA_F32(A, Bv, C) \
    __builtin_amdgcn_wmma_f32_16x16x4_f32(false, (A), false, (Bv), (short)0, (C), false, false)

__device__ __forceinline__ float fast_exp(float v) {
    return __builtin_amdgcn_exp2f(v * 1.4426950408889634f);
}
__device__ __forceinline__ float fast_sigmoid(float v) {
    return __builtin_amdgcn_rcpf(1.0f + fast_exp(-v));
}
__device__ __forceinline__ float fast_tanh(float v) {
#if __has_builtin(__builtin_amdgcn_tanhf)
    return __builtin_amdgcn_tanhf(v);
#else
    return 1.0f - 2.0f * __builtin_amdgcn_rcpf(1.0f + fast_exp(2.0f * v));
#endif
}

__global__ __launch_bounds__(32)
__attribute__((amdgpu_num_vgpr(256)))
void gru2_fused(
    const int*   __restrict__ x,
    const float* __restrict__ W0, const float* __restrict__ U0,
    const float* __restrict__ b0i, const float* __restrict__ b0r,
    const float* __restrict__ W1, const float* __restrict__ U1,
    const float* __restrict__ b1i, const float* __restrict__ b1r,
    const float* __restrict__ Wd, const float* __restrict__ bd,
    float* __restrict__ out)
{
    extern __shared__ char smem[];
    float* W0s  = (float*)(smem + kW0Off);   // [256][60], b0i pre-added
    int*   xIdx = (int*)  (smem + kXOff);    // [16][129]
    float* h0s  = (float*)(smem + kH0Off);   // [16][22]
    float* h1s  = (float*)(smem + kH1Off);   // [16][22]
    float* recs = (float*)(smem + kRecOff);  // [16][66]
    float* xws  = (float*)(smem + kXwOff);   // [16][66]
    v2f*   w1L  = (v2f*)  (smem + kW1fOff);  // [20][32] per-lane W1 B-frags
    v2f*   u1L  = (v2f*)  (smem + kU1fOff);  // [20][32] per-lane U1 B-frags

    const int lane = threadIdx.x;     // 0..31 (wave32)
    const int hf   = lane >> 4;       // lane half
    const int n    = lane & 15;
    const int b0   = blockIdx.x * 16; // batch tile base

    // ---- stage W0 (+b0i) into LDS: 3840 float4, coalesced ----
    {
        const float4* wsrc = (const float4*)W0;
        const float4* bi4  = (const float4*)b0i;   // 15 float4 (60 % 4 == 0)
        float4* wdst = (float4*)W0s;
        for (int f = lane; f < kV * kH3 / 4; f += 32) {
            float4 w = wsrc[f];
            float4 b = bi4[f % (kH3 / 4)];
            w.x += b.x; w.y += b.y; w.z += b.z; w.w += b.w;
            wdst[f] = w;
        }
    }
    // ---- stage token ids (padded rows): int4 global loads, scalar LDS stores ----
    {
        const int4* xsrc = (const int4*)(x + b0 * kT);   // 512 int4, coalesced
        #pragma unroll
        for (int k = 0; k < 16; ++k) {
            const int f = lane + 32 * k;      // int4 index; row = f/32 (no row crossing)
            const int4 v = xsrc[f];
            const int m = f >> 5;
            const int tb = (f & 31) * 4;
            int* dst = xIdx + m * kXP + tb;
            dst[0] = v.x; dst[1] = v.y; dst[2] = v.z; dst[3] = v.w;
        }
    }

    // ---- preload weights as WMMA B-fragments ----
    // B tile (N-tile i, K-step k): lane column c = 16*i + n,
    //   VGPR0 = U[4k + 2*hf][c], VGPR1 = U[4k + 2*hf + 1][c]
    // U0 frags stay in VGPRs (layer-0 starts each step's critical chain);
    // W1/U1 frags staged to LDS (keeps total VGPR demand < 256: no spills).
    v2f U0f[kNT][kNK];
    float b0r4[kNT], b1i4[kNT], b1r4[kNT];   // per-lane column biases
    #pragma unroll
    for (int i = 0; i < kNT; ++i) {
        const int c = 16 * i + n;
        const bool cv = (c < kH3);
        b0r4[i] = cv ? b0r[c] : 0.0f;
        b1i4[i] = cv ? b1i[c] : 0.0f;
        b1r4[i] = cv ? b1r[c] : 0.0f;
        #pragma unroll
        for (int k = 0; k < kNK; ++k) {
            const int kb = 4 * k + 2 * hf;  // kb+1 <= 19
            v2f u0, w1, u1;
            u0.x = cv ? U0[kb * kH3 + c] : 0.0f;
            u0.y = cv ? U0[(kb + 1) * kH3 + c] : 0.0f;
            w1.x = cv ? W1[kb * kH3 + c] : 0.0f;
            w1.y = cv ? W1[(kb + 1) * kH3 + c] : 0.0f;
            u1.x = cv ? U1[kb * kH3 + c] : 0.0f;
            u1.y = cv ? U1[(kb + 1) * kH3 + c] : 0.0f;
            U0f[i][k] = u0;
            w1L[(i * kNK + k) * 32 + lane] = w1;   // lane-major: conflict-free b64
            u1L[(i * kNK + k) * 32 + lane] = u1;
        }
    }

    // ---- zero hidden state (16*22 = 352 = 11*32 each) ----
    #pragma unroll
    for (int e = 0; e < 11; ++e) {
        const int flat = lane + 32 * e;
        h0s[flat] = 0.0f;
        h1s[flat] = 0.0f;
    }
    __syncthreads();

    // per-lane gate-phase bases: this lane owns batch row m = n, j = hf + 2e
    const int*   xrow = xIdx + n * kXP;
    float*       h0p  = h0s + n * kHP;
    float*       h1p  = h1s + n * kHP;
    const float* rp   = recs + n * kRP;
    const float* xp   = xws + n * kRP;

    for (int t = 0; t < kT; ++t) {
        // ================= layer 0: rec0 = h0 @ U0 (+ b0r at dump) =================
        {
            v2f a[kNK];
            #pragma unroll
            for (int k = 0; k < kNK; ++k) {            // A frag: m = n, K = kb, kb+1
                const int kb = 4 * k + 2 * hf;
                a[k].x = h0p[kb];
                a[k].y = h0p[kb + 1];
            }
            v8f r0 = {}, r1 = {}, r2 = {}, r3 = {};    // C = inline 0 on first WMMA
            #pragma unroll
            for (int k = 0; k < kNK; ++k) {
                r0 = WMMA_F32(a[k], U0f[0][k], r0);
                r1 = WMMA_F32(a[k], U0f[1][k], r1);
                r2 = WMMA_F32(a[k], U0f[2][k], r2);
                r3 = WMMA_F32(a[k], U0f[3][k], r3);
            }
            #pragma unroll
            for (int r = 0; r < 8; ++r) {              // D: (M = r + 8*hf, N = n)
                float* d = recs + (r + 8 * hf) * kRP + n;
                d[0]  = r0[r] + b0r4[0];
                d[16] = r1[r] + b0r4[1];
                d[32] = r2[r] + b0r4[2];
                d[48] = r3[r] + b0r4[3];
            }
        }
        __syncthreads();

        // ---- layer-0 gates: lane handles m = n, j = hf + 2e (one token row/lane) ----
        {
            const float* wrow = W0s + xrow[t] * kH3;   // b0i pre-folded
            #pragma unroll
            for (int e = 0; e < 10; ++e) {
                const int j = hf + 2 * e;
                const float z  = fast_sigmoid(wrow[j] + rp[j]);
                const float rg = fast_sigmoid(wrow[kH + j] + rp[kH + j]);
                const float hh = fast_tanh(wrow[2 * kH + j] + rg * rp[2 * kH + j]);
                h0p[j] = z * h0p[j] + (1.0f - z) * hh;
            }
        }
        __syncthreads();

        // ================= layer 1: xw1 = h0_new @ W1 (+ b1i at dump) =================
        {
            v2f a[kNK];
            #pragma unroll
            for (int k = 0; k < kNK; ++k) {
                const int kb = 4 * k + 2 * hf;
                a[k].x = h0p[kb];
                a[k].y = h0p[kb + 1];
            }
            v8f r0 = {}, r1 = {}, r2 = {}, r3 = {};
            #pragma unroll
            for (int k = 0; k < kNK; ++k) {            // weight frags streamed from LDS
                const v2f w0 = w1L[(0 * kNK + k) * 32 + lane];
                const v2f w1f = w1L[(1 * kNK + k) * 32 + lane];
                const v2f w2 = w1L[(2 * kNK + k) * 32 + lane];
                const v2f w3 = w1L[(3 * kNK + k) * 32 + lane];
                r0 = WMMA_F32(a[k], w0, r0);
                r1 = WMMA_F32(a[k], w1f, r1);
                r2 = WMMA_F32(a[k], w2, r2);
                r3 = WMMA_F32(a[k], w3, r3);
            }
            #pragma unroll
            for (int r = 0; r < 8; ++r) {
                float* d = xws + (r + 8 * hf) * kRP + n;
                d[0]  = r0[r] + b1i4[0];
                d[16] = r1[r] + b1i4[1];
                d[32] = r2[r] + b1i4[2];
                d[48] = r3[r] + b1i4[3];
            }
        }
        // ================= layer 1: rec1 = h1 @ U1 (+ b1r at dump) =================
        {
            v2f a[kNK];
            #pragma unroll
            for (int k = 0; k < kNK; ++k) {
                const int kb = 4 * k + 2 * hf;
                a[k].x = h1p[kb];
                a[k].y = h1p[kb + 1];
            }
            v8f r0 = {}, r1 = {}, r2 = {}, r3 = {};
            #pragma unroll
            for (int k = 0; k < kNK; ++k) {
                const v2f u0 = u1L[(0 * kNK + k) * 32 + lane];
                const v2f u1f = u1L[(1 * kNK + k) * 32 + lane];
                const v2f u2 = u1L[(2 * kNK + k) * 32 + lane];
                const v2f u3 = u1L[(3 * kNK + k) * 32 + lane];
                r0 = WMMA_F32(a[k], u0, r0);
                r1 = WMMA_F32(a[k], u1f, r1);
                r2 = WMMA_F32(a[k], u2, r2);
                r3 = WMMA_F32(a[k], u3, r3);
            }
            #pragma unroll
            for (int r = 0; r < 8; ++r) {
                float* d = recs + (r + 8 * hf) * kRP + n;
                d[0]  = r0[r] + b1r4[0];
                d[16] = r1[r] + b1r4[1];
                d[32] = r2[r] + b1r4[2];
                d[48] = r3[r] + b1r4[3];
            }
        }
        __syncthreads();

        // ---- layer-1 gates ----
        #pragma unroll
        for (int e = 0; e < 10; ++e) {
            const int j = hf + 2 * e;
            const float z  = fast_sigmoid(xp[j] + rp[j]);
            const float rg = fast_sigmoid(xp[kH + j] + rp[kH + j]);
            const float hh = fast_tanh(xp[2 * kH + j] + rg * rp[2 * kH + j]);
            h1p[j] = z * h1p[j] + (1.0f - z) * hh;
        }
        __syncthreads();
    }

    // ---- dense head + softmax: lane m < 16 handles batch row b0+m ----
    if (lane < 16) {
        float h[kH];
        #pragma unroll
        for (int k = 0; k < kH; ++k) h[k] = h1s[lane * kHP + k];
        float v[kL];
        float mx = -3.4e38f;
        #pragma unroll
        for (int j = 0; j < kL; ++j) {
            float s = bd[j];
            #pragma unroll
            for (int k = 0; k < kH; ++k) s = fmaf(h[k], Wd[k * kL + j], s);
            v[j] = s;
            mx = fmaxf(mx, s);
        }
        float sum = 0.0f;
        #pragma unroll
        for (int j = 0; j < kL; ++j) { v[j] = expf(v[j] - mx); sum += v[j]; }
        const float inv = 1.0f / sum;
        #pragma unroll
        for (int j = 0; j < kL; ++j) out[(b0 + lane) * kL + j] = v[j] * inv;
    }
}

extern "C" void kernel_launch(void* const* d_in, const int* in_sizes, int n_in,
                              void* d_out, int out_size, void* d_ws, size_t ws_size,
                              hipStream_t stream) {
    (void)in_sizes; (void)n_in; (void)d_ws; (void)ws_size; (void)out_size;
    const int*   x   = (const int*)  d_in[0];
    const float* W0  = (const float*)d_in[1];
    const float* U0  = (const float*)d_in[2];
    const float* b0i = (const float*)d_in[3];
    const float* b0r = (const float*)d_in[4];
    const float* W1  = (const float*)d_in[5];
    const float* U1  = (const float*)d_in[6];
    const float* b1i = (const float*)d_in[7];
    const float* b1r = (const float*)d_in[8];
    const float* Wd  = (const float*)d_in[9];
    const float* bd  = (const float*)d_in[10];
    // d_in[11] = drop_rate (identity dropout in reference; unused)
    float* out = (float*)d_out;

    // allow > 64 KB dynamic LDS (host-side attribute; graph-capture safe)
    (void)hipFuncSetAttribute((const void*)gru2_fused,
                              hipFuncAttributeMaxDynamicSharedMemorySize, kLdsBytes);

    dim3 grid(kB / 16);   // 128 batch tiles of 16 rows, one wave32 each
    dim3 block(32);
    gru2_fused<<<grid, block, kLdsBytes, stream>>>(x, W0, U0, b0i, b0r,
                                                   W1, U1, b1i, b1r, Wd, bd, out);
}